// SwitchMoE_13185549598920
// MI455X (gfx1250) — compile-verified
//
#include <hip/hip_runtime.h>
#include <hip/hip_bf16.h>

#define E_   8
#define DIM_ 128
#define HID_ 512
#define OUT_ 128
#define BB_  16
#define HH_  64
#define WW_  64
#define NN_  (HH_ * WW_)
#define EPS_ 1e-6f
#define GATE_CHUNKS_ 16   // blocks per batch for the gate scan (256 blocks total)

typedef __attribute__((ext_vector_type(2))) float v2f;
typedef __attribute__((ext_vector_type(8))) float v8f;

// ---------------------------------------------------------------------------
// Zero the whole output (only rows n<8 per batch get overwritten later).
// 33.5 MB of 128-bit stores -> ~1.5us at 23.3 TB/s.
// ---------------------------------------------------------------------------
__global__ void k_zero(float4* __restrict__ out, int n4) {
    int i = blockIdx.x * blockDim.x + threadIdx.x;
    if (i < n4) out[i] = make_float4(0.f, 0.f, 0.f, 0.f);
}

// ---------------------------------------------------------------------------
// Init the small gate workspace (hit flags must start at 0 every call; the
// harness does not re-poison d_ws between replays).
// ---------------------------------------------------------------------------
__global__ void k_init(float* __restrict__ ws_small) {
    ws_small[threadIdx.x] = 0.f;   // 384 floats: g, hit, gate
}

// ---------------------------------------------------------------------------
// Gate scan, split 16 ways per batch for bandwidth: grid (chunk, batch),
// 1 token per thread, float4 row loads. For every token: 8 logits,
// first-index argmax -> hit flag (idempotent 1.0f store); tokens n<8 also
// store softmax prob of expert 0. Reads all of x once (33.5 MB).
// ---------------------------------------------------------------------------
__global__ void k_gate_scan(const float* __restrict__ x,
                            const float* __restrict__ wg,
                            const float* __restrict__ bg,
                            float* __restrict__ ws_g,
                            float* __restrict__ ws_hit) {
    __shared__ float s_wg[DIM_ * E_];
    __shared__ float s_bg[E_];
    const int b   = blockIdx.y;
    const int tid = threadIdx.x;
    const int m   = blockIdx.x * blockDim.x + tid;   // token 0..4095
    for (int i = tid; i < DIM_ * E_; i += blockDim.x) s_wg[i] = wg[i];
    if (tid < E_) s_bg[tid] = bg[tid];
    __syncthreads();

    const float4* xr = (const float4*)(x + ((size_t)b * NN_ + m) * DIM_);
    float l[E_];
    #pragma unroll
    for (int e = 0; e < E_; ++e) l[e] = s_bg[e];
    #pragma unroll 4
    for (int d4 = 0; d4 < DIM_ / 4; ++d4) {
        const float4 xv = xr[d4];
        const int d = d4 * 4;
        #pragma unroll
        for (int e = 0; e < E_; ++e) {
            float acc = l[e];
            acc = fmaf(xv.x, s_wg[(d + 0) * E_ + e], acc);
            acc = fmaf(xv.y, s_wg[(d + 1) * E_ + e], acc);
            acc = fmaf(xv.z, s_wg[(d + 2) * E_ + e], acc);
            acc = fmaf(xv.w, s_wg[(d + 3) * E_ + e], acc);
            l[e] = acc;
        }
    }
    // first-index argmax (strict >), matches jnp.argmax tie-break
    int arg = 0; float mx = l[0];
    #pragma unroll
    for (int e = 1; e < E_; ++e) if (l[e] > mx) { mx = l[e]; arg = e; }
    ws_hit[b * E_ + arg] = 1.0f;   // benign race: every writer stores 1.0f
    if (m < E_) {
        float ssum = 0.f;
        #pragma unroll
        for (int e = 0; e < E_; ++e) ssum += expf(l[e] - mx);
        ws_g[b * E_ + m] = expf(l[0] - mx) / ssum;   // softmax prob of expert 0
    }
}

// ---------------------------------------------------------------------------
// gate[b,n] = g*hit / (sum_b g*hit + eps) * capacity(16)
// ---------------------------------------------------------------------------
__global__ void k_gate_fin(const float* __restrict__ ws_g,
                           const float* __restrict__ ws_hit,
                           float* __restrict__ ws_gate) {
    __shared__ float sm[BB_ * E_];
    const int t = threadIdx.x;          // 128 = BB_*E_
    const float m = ws_g[t] * ws_hit[t];
    sm[t] = m;
    __syncthreads();
    const int n = t & (E_ - 1);
    float denom = EPS_;
    #pragma unroll
    for (int b = 0; b < BB_; ++b) denom += sm[b * E_ + n];
    ws_gate[t] = m / denom * 16.0f;
}

// ---------------------------------------------------------------------------
// fc1 for expert 0 over the 288 live rows (16 batches x {y=0,1} x {x=0..8}).
// One wave per 16x16 output tile; K=128 as 32x V_WMMA_F32_16X16X4_F32.
// A frag:  v0/v1[lane] = A[lane%16][2*(lane/16) + {0,1}]
// B frag:  v0/v1[lane] = B[2*(lane/16) + {0,1}][lane%16]
// C frag:  v_i[lane]   = C[i + 8*(lane/16)][lane%16]
// ---------------------------------------------------------------------------
__global__ void k_fc1(const float* __restrict__ x,
                      const float* __restrict__ W1,   // (DIM, HID) expert 0
                      const float* __restrict__ B1,   // (HID) expert 0
                      float* __restrict__ h1) {       // (288, HID)
    const int lane = threadIdx.x;
    const int mh   = lane & 15;
    const int grp  = lane >> 4;
    const int tm   = blockIdx.y;        // 0..17
    const int tn   = blockIdx.x;        // 0..31
    const int r    = tm * 16 + mh;      // live row 0..287
    const int b    = r / 18;
    const int rem  = r % 18;
    const int y    = rem / 9;
    const int xc   = rem % 9;
    const float* arow = x + ((size_t)b * NN_ + y * WW_ + xc) * DIM_;
    const int n = tn * 16 + mh;

    v8f acc = {};
    #pragma unroll 4
    for (int k = 0; k < DIM_; k += 4) {
        const int ka = k + 2 * grp;
        v2f a, bv;
        a[0]  = arow[ka];
        a[1]  = arow[ka + 1];
        bv[0] = W1[(size_t)ka * HID_ + n];
        bv[1] = W1[(size_t)(ka + 1) * HID_ + n];
        acc = __builtin_amdgcn_wmma_f32_16x16x4_f32(false, a, false, bv,
                                                    (short)0, acc, false, false);
    }
    const float bias = B1[n];
    #pragma unroll
    for (int i = 0; i < 8; ++i) {
        const int mrow = tm * 16 + i + 8 * grp;
        h1[(size_t)mrow * HID_ + n] = acc[i] + bias;
    }
}

// ---------------------------------------------------------------------------
// Depthwise 3x3 (only taps landing on live output row y=0, cols 0..7) + bias
// + exact GELU. h1 rows are [b][y in 0..1][x in 0..8].
// ---------------------------------------------------------------------------
__global__ void k_dwgelu(const float* __restrict__ h1,
                         const float* __restrict__ Wd,   // (3,3,1,HID) expert 0
                         const float* __restrict__ Bd,   // (HID) expert 0
                         float* __restrict__ act) {      // (128, HID)
    const int rc = blockIdx.x;          // 0..127 -> (b, c)
    const int b  = rc >> 3;
    const int c  = rc & 7;
    for (int hh = threadIdx.x; hh < HID_; hh += blockDim.x) {
        float acc = Bd[hh];
        #pragma unroll
        for (int ky = 1; ky < 3; ++ky) {     // ky=0 -> y'=-1 is padding
            const int yy = ky - 1;           // source row 0 or 1
            #pragma unroll
            for (int kx = 0; kx < 3; ++kx) {
                const int xx = c + kx - 1;   // -1..8 ; -1 is padding, 8 is real
                if (xx >= 0) {
                    const float hv = h1[((size_t)(b * 18 + yy * 9 + xx)) * HID_ + hh];
                    const float wv = Wd[(ky * 3 + kx) * HID_ + hh];
                    acc = fmaf(hv, wv, acc);
                }
            }
        }
        // exact GELU: 0.5*v*(1+erf(v/sqrt(2)))
        act[(size_t)rc * HID_ + hh] =
            0.5f * acc * (1.0f + erff(acc * 0.7071067811865475f));
    }
}

// ---------------------------------------------------------------------------
// fc2 (M=128 live rows, K=512, N=128) fused with +B2, gate scale, and scatter
// into d_out rows (b, n=c<8). 128x V_WMMA_F32_16X16X4_F32 per wave.
// ---------------------------------------------------------------------------
__global__ void k_fc2(const float* __restrict__ act,
                      const float* __restrict__ W2,     // (HID, OUT) expert 0
                      const float* __restrict__ B2,     // (OUT) expert 0
                      const float* __restrict__ ws_gate,
                      float* __restrict__ out) {
    const int lane = threadIdx.x;
    const int mh   = lane & 15;
    const int grp  = lane >> 4;
    const int tm   = blockIdx.y;        // 0..7
    const int tn   = blockIdx.x;        // 0..7
    const int r    = tm * 16 + mh;
    const float* arow = act + (size_t)r * HID_;
    const int o = tn * 16 + mh;

    v8f acc = {};
    #pragma unroll 4
    for (int k = 0; k < HID_; k += 4) {
        const int ka = k + 2 * grp;
        v2f a, bv;
        a[0]  = arow[ka];
        a[1]  = arow[ka + 1];
        bv[0] = W2[(size_t)ka * OUT_ + o];
        bv[1] = W2[(size_t)(ka + 1) * OUT_ + o];
        acc = __builtin_amdgcn_wmma_f32_16x16x4_f32(false, a, false, bv,
                                                    (short)0, acc, false, false);
    }
    const float bias = B2[o];
    #pragma unroll
    for (int i = 0; i < 8; ++i) {
        const int mrow = tm * 16 + i + 8 * grp;   // live row 0..127
        const int b = mrow >> 3;
        const int c = mrow & 7;
        const float gt = ws_gate[mrow];
        out[((size_t)b * NN_ + c) * OUT_ + o] = gt * (acc[i] + bias);
    }
}

// ---------------------------------------------------------------------------
extern "C" void kernel_launch(void* const* d_in, const int* in_sizes, int n_in,
                              void* d_out, int out_size, void* d_ws, size_t ws_size,
                              hipStream_t stream) {
    (void)in_sizes; (void)n_in; (void)out_size; (void)ws_size;
    const float* x  = (const float*)d_in[0];
    // d_in[1]=H, d_in[2]=W (python ints; fixed at 64 per the reference)
    const float* wg = (const float*)d_in[3];
    const float* bg = (const float*)d_in[4];
    const float* W1 = (const float*)d_in[5];   // (E, DIM, HID) -> expert 0 slice
    const float* B1 = (const float*)d_in[6];
    const float* Wd = (const float*)d_in[7];   // (E, 3, 3, 1, HID)
    const float* Bd = (const float*)d_in[8];
    const float* W2 = (const float*)d_in[9];   // (E, HID, OUT)
    const float* B2 = (const float*)d_in[10];
    float* out = (float*)d_out;
    float* ws  = (float*)d_ws;

    float* ws_g    = ws;                    // 128
    float* ws_hit  = ws + 128;              // 128
    float* ws_gate = ws + 256;              // 128
    float* ws_h1   = ws + 384;              // 288 * 512
    float* ws_act  = ws_h1 + 288 * HID_;    // 128 * 512

    const int n4 = BB_ * NN_ * OUT_ / 4;
    k_zero<<<(n4 + 255) / 256, 256, 0, stream>>>((float4*)out, n4);
    k_init<<<1, 384, 0, stream>>>(ws);
    k_gate_scan<<<dim3(GATE_CHUNKS_, BB_), NN_ / GATE_CHUNKS_, 0, stream>>>(
        x, wg, bg, ws_g, ws_hit);
    k_gate_fin<<<1, BB_ * E_, 0, stream>>>(ws_g, ws_hit, ws_gate);
    k_fc1<<<dim3(HID_ / 16, 288 / 16), 32, 0, stream>>>(x, W1, B1, ws_h1);
    k_dwgelu<<<BB_ * 8, 256, 0, stream>>>(ws_h1, Wd, Bd, ws_act);
    k_fc2<<<dim3(OUT_ / 16, 128 / 16), 32, 0, stream>>>(ws_act, W2, B2, ws_gate, out);
}